// spixel_upsample2d_78752520339850
// MI455X (gfx1250) — compile-verified
//
#include <hip/hip_runtime.h>

// Problem constants (fixed shapes from the reference).
constexpr int Bn  = 4;        // batch
constexpr int Cn  = 2;        // cv channels
constexpr int Hn  = 128;      // low-res H
constexpr int Wn  = 256;      // low-res W
constexpr int NCH = Cn * 9;   // 18 sp channels
constexpr int H4  = 4 * Hn;   // 512
constexpr int W4  = 4 * Wn;   // 1024

// Tile: one low-res row (4 output rows) x 64 low-res cols (256 output cols).
#define BDIMX 64
#define BDIMY 4

#if defined(__has_builtin)
#  if __has_builtin(__builtin_amdgcn_global_load_async_to_lds_b32)
#    define HAVE_ASYNC_LDS 1
#  endif
#endif
#ifndef HAVE_ASYNC_LDS
#  define HAVE_ASYNC_LDS 0
#  warning "gfx1250 async global->LDS builtin not available; using ds_store fallback"
#endif

// Address-space-qualified int pointers for the gfx1250 async builtin
// (AS1 global source == '__device__ int *', AS3 LDS dest == '__shared__ int *').
typedef __attribute__((address_space(1))) int as1_int;
typedef __attribute__((address_space(3))) int as3_int;

// True 4-wide float vector so __builtin_nontemporal_load/store apply.
typedef float v4f __attribute__((ext_vector_type(4)));

static __device__ __forceinline__ void wait_async_zero() {
#if defined(__has_builtin)
#  if __has_builtin(__builtin_amdgcn_s_wait_asynccnt)
  __builtin_amdgcn_s_wait_asynccnt(0);
  return;
#  endif
#endif
  asm volatile("s_wait_asynccnt 0" ::: "memory");
}

__global__ __launch_bounds__(BDIMX * BDIMY)
void spixel_upsample2d_kernel(const float* __restrict__ cv,
                              const float* __restrict__ sp,
                              float* __restrict__ out) {
  // LDS halo tile of cv: 2 channels x 3 rows x 66 cols (pad stride 72).
  __shared__ float scv[Cn][3][72];

  const int tx  = threadIdx.x;               // low-res col within tile, 0..63
  const int a   = threadIdx.y;               // sub-row, 0..3
  const int tid = a * BDIMX + tx;            // 0..255

  const int xtile = blockIdx.x;              // 0..3
  const int hb    = blockIdx.y;              // low-res row, 0..127
  const int b     = blockIdx.z;              // batch, 0..3

  const int w0 = xtile * BDIMX;              // low-res col base of tile

  // ---- Stage cv halo into LDS (async global->LDS for in-range, ds zero for pad)
  constexpr int NSTAGE = Cn * 3 * 66;        // 396 elements
  for (int i = tid; i < NSTAGE; i += BDIMX * BDIMY) {
    const int c   = i / (3 * 66);
    const int rem = i % (3 * 66);
    const int r   = rem / 66;                // 0..2 -> row hb-1+r
    const int col = rem % 66;                // 0..65 -> col w0-1+col
    const int hh  = hb - 1 + r;
    const int ww  = w0 - 1 + col;
    float* ldst = &scv[c][r][col];
    if (hh >= 0 && hh < Hn && ww >= 0 && ww < Wn) {
      const float* gsrc = cv + (((b * Cn + c) * Hn + hh) * Wn + ww);
#if HAVE_ASYNC_LDS
      __builtin_amdgcn_global_load_async_to_lds_b32(
          (as1_int*)(void*)gsrc, (as3_int*)(void*)ldst, 0, 0);
#else
      *ldst = *gsrc;
#endif
    } else {
      *ldst = 0.0f;                          // zero padding (ds_store)
    }
  }
#if HAVE_ASYNC_LDS
  wait_async_zero();                          // my wave's async copies done
#endif
  __syncthreads();                            // everyone's copies visible

  // ---- Gather the 18 patch scalars for this thread's low-res cell (w0+tx)
  float p[NCH];
#pragma unroll
  for (int c = 0; c < Cn; ++c)
#pragma unroll
    for (int ky = 0; ky < 3; ++ky)
#pragma unroll
      for (int kx = 0; kx < 3; ++kx)
        p[c * 9 + ky * 3 + kx] = scv[c][ky][tx + kx];

  // ---- Stream 18 sp planes with non-temporal b128 loads: sp has zero reuse,
  //      keep the 151 MB stream from rinsing L2 so cv halo rows stay resident.
  const int  Y     = hb * 4 + a;
  const int  X     = w0 * 4 + tx * 4;
  const long plane = (long)H4 * W4;
  const float* spb = sp + (long)b * NCH * plane + (long)Y * W4 + X;

  v4f sum = (v4f)(0.0f);
#pragma unroll
  for (int ch = 0; ch < NCH; ++ch) {
    const v4f s = __builtin_nontemporal_load((const v4f*)(spb + (long)ch * plane));
    sum = s * p[ch] + sum;                    // vector FMA (contracted at -O3)
  }

  __builtin_nontemporal_store(sum, (v4f*)(out + (long)b * plane + (long)Y * W4 + X));
}

extern "C" void kernel_launch(void* const* d_in, const int* in_sizes, int n_in,
                              void* d_out, int out_size, void* d_ws, size_t ws_size,
                              hipStream_t stream) {
  (void)in_sizes; (void)n_in; (void)out_size; (void)d_ws; (void)ws_size;
  const float* cv = (const float*)d_in[0];   // (4, 2, 1, 128, 256) f32
  const float* sp = (const float*)d_in[1];   // (4, 18, 512, 1024) f32
  float* out = (float*)d_out;                // (4, 1, 512, 1024) f32

  dim3 block(BDIMX, BDIMY, 1);
  dim3 grid(W4 / (BDIMX * 4), Hn, Bn);       // (4, 128, 4)
  spixel_upsample2d_kernel<<<grid, block, 0, stream>>>(cv, sp, out);
}